// ReBertCrf_38062000177708
// MI455X (gfx1250) — compile-verified
//
#include <hip/hip_runtime.h>
#include <hip/hip_bf16.h>

// Problem constants (from reference): BS=16, S=256, H=1024, T=32.
// M = BS*S = 4096 flattened rows of entities_embeddings.

typedef __attribute__((ext_vector_type(2))) float v2f;
typedef __attribute__((ext_vector_type(4))) float v4f;
typedef __attribute__((ext_vector_type(8))) float v8f;

#define BSZ 16
#define SEQ 256
#define HID 1024
#define TAG 32
#define MTOT (BSZ * SEQ)   // 4096

// ---------------------------------------------------------------------------
// Kernel 1: ps[b,t] = seq[b,:] . W3[t,:] + bias[t]    (16 x 32 output)
// W3 = W[:, 2H:3H], W row-major (32, 3072).
// ---------------------------------------------------------------------------
__global__ __launch_bounds__(32) void ps_kernel(const float* __restrict__ seq,
                                                const float* __restrict__ W,
                                                const float* __restrict__ bias,
                                                float* __restrict__ ps) {
  const int b = blockIdx.x;   // 0..15
  const int t = threadIdx.x;  // 0..31
  const v4f* sv = (const v4f*)(seq + (size_t)b * HID);
  const v4f* wv = (const v4f*)(W + (size_t)t * (3 * HID) + 2 * HID);
  float acc = 0.0f;
  #pragma unroll 4
  for (int h = 0; h < HID / 4; ++h) {
    v4f s = sv[h];
    v4f w = wv[h];
    acc += s.x * w.x + s.y * w.y + s.z * w.z + s.w * w.w;
  }
  ps[b * TAG + t] = acc + bias[t];
}

// ---------------------------------------------------------------------------
// Kernel 2: WMMA fp32 GEMM.
//   P[:, 0:32]  = E (4096x1024) . W1^T   (+ ps broadcast)  -> p1ps
//   P[:, 32:64] = E . W2^T                                 -> p2
// One wave = one 16-row M tile x all 64 N columns (4 16x16 tiles).
// v_wmma_f32_16x16x4_f32:  A 16x4 (2 VGPR/lane), B 4x16 (2 VGPR/lane),
// C/D 16x16 f32 (8 VGPR). Lane layout (ISA 7.12.2):
//   A: lane L holds A[L%16, 2*(L/16)+j] in vgpr j
//   B: lane L holds B[2*(L/16)+j, L%16] in vgpr j  (B = W'^T tile, so this
//      is W'[n0 + L%16, k + 2*(L/16)+j] -> contiguous v2f load per lane)
//   D: lane L vgpr r holds D[r + 8*(L/16), L%16]
// ---------------------------------------------------------------------------
__global__ __launch_bounds__(128) void gemm_wmma_kernel(
    const float* __restrict__ E,    // (4096, 1024)
    const float* __restrict__ W,    // (32, 3072)  [W1 | W2 | W3]
    const float* __restrict__ ps,   // (16, 32)
    float* __restrict__ p1ps,       // (4096, 32)
    float* __restrict__ p2) {       // (4096, 32)
  const int lane  = threadIdx.x & 31;
  const int wave  = threadIdx.x >> 5;          // 0..3
  const int lmod  = lane & 15;
  const int lhalf = lane >> 4;                 // 0 or 1
  const int m0    = (blockIdx.x * 4 + wave) * 16;   // < 4096

  // A fragment source: row (m0 + lmod), cols k + 2*lhalf, k + 2*lhalf + 1
  const float* Arow = E + (size_t)(m0 + lmod) * HID + 2 * lhalf;

  // B fragment sources: W' rows are [W1 rows 0..31 ; W2 rows 0..31]
  const float* B0 = W + (size_t)lmod * (3 * HID) + 2 * lhalf;        // W1 t=0..15
  const float* B1 = W + (size_t)(lmod + 16) * (3 * HID) + 2 * lhalf; // W1 t=16..31
  const float* B2 = B0 + HID;                                        // W2 t=0..15
  const float* B3 = B1 + HID;                                        // W2 t=16..31

  v8f a0 = {}, a1 = {}, a2 = {}, a3 = {};
  #pragma unroll 4
  for (int k = 0; k < HID; k += 4) {
    v2f av = *(const v2f*)(Arow + k);
    v2f b0 = *(const v2f*)(B0 + k);
    v2f b1 = *(const v2f*)(B1 + k);
    v2f b2 = *(const v2f*)(B2 + k);
    v2f b3 = *(const v2f*)(B3 + k);
    a0 = __builtin_amdgcn_wmma_f32_16x16x4_f32(false, av, false, b0, (short)0, a0, false, false);
    a1 = __builtin_amdgcn_wmma_f32_16x16x4_f32(false, av, false, b1, (short)0, a1, false, false);
    a2 = __builtin_amdgcn_wmma_f32_16x16x4_f32(false, av, false, b2, (short)0, a2, false, false);
    a3 = __builtin_amdgcn_wmma_f32_16x16x4_f32(false, av, false, b3, (short)0, a3, false, false);
  }

  // Each 16-row tile stays inside one batch b (256 % 16 == 0).
  const int b = m0 >> 8;
  const float psLo = ps[b * TAG + lmod];
  const float psHi = ps[b * TAG + 16 + lmod];
  #pragma unroll
  for (int r = 0; r < 8; ++r) {
    const int row = m0 + r + 8 * lhalf;
    p1ps[(size_t)row * TAG + lmod]      = a0[r] + psLo;
    p1ps[(size_t)row * TAG + 16 + lmod] = a1[r] + psHi;
    p2  [(size_t)row * TAG + lmod]      = a2[r];
    p2  [(size_t)row * TAG + 16 + lmod] = a3[r];
  }
}

// ---------------------------------------------------------------------------
// Kernel 3: broadcast add (HBM-bound: 128 MB of NT stores).
//   out[b,i,j,t] = p1ps[b,i,t] + p2[b,j,t]
// One block per (b,i). Each (b,i) slice is 256*32 floats = 2048 float4s;
// 256 threads x 8 iterations, consecutive tids -> consecutive float4 stores.
// float4 index f:  j = f>>3, t4 = f&7; p2 float4 index = f; p1 index = f&7.
// ---------------------------------------------------------------------------
__global__ __launch_bounds__(256) void bcast_kernel(const float* __restrict__ p1ps,
                                                    const float* __restrict__ p2,
                                                    float* __restrict__ out) {
  const int bi  = blockIdx.x;       // b*256 + i
  const int b   = bi >> 8;
  const int tid = threadIdx.x;
  const v4f* p1v = (const v4f*)(p1ps + (size_t)bi * TAG);
  const v4f* p2v = (const v4f*)(p2 + (size_t)b * (SEQ * TAG));
  v4f* outv = (v4f*)(out + (size_t)bi * (SEQ * TAG));
  const v4f a = p1v[tid & 7];
  #pragma unroll
  for (int it = 0; it < 8; ++it) {
    const int f = tid + 256 * it;
    v4f v = p2v[f];
    v += a;
    __builtin_nontemporal_store(v, outv + f);  // output never re-read: stream past L2
  }
}

// ---------------------------------------------------------------------------
extern "C" void kernel_launch(void* const* d_in, const int* in_sizes, int n_in,
                              void* d_out, int out_size, void* d_ws, size_t ws_size,
                              hipStream_t stream) {
  const float* seq  = (const float*)d_in[0];  // (16, 1024)
  const float* ents = (const float*)d_in[1];  // (16, 256, 1024)
  const float* W    = (const float*)d_in[2];  // (32, 3072)
  const float* bias = (const float*)d_in[3];  // (32,)
  // d_in[4] = entities_tags: unused by the reference output.
  float* out = (float*)d_out;                 // (16, 256, 256, 32)

  float* ws   = (float*)d_ws;
  float* ps   = ws;                           // 16*32        = 512 floats
  float* p1ps = ws + 512;                     // 4096*32      = 131072 floats
  float* p2   = p1ps + (size_t)MTOT * TAG;    // 4096*32      = 131072 floats
  // total workspace: ~1.03 MB

  ps_kernel<<<BSZ, 32, 0, stream>>>(seq, W, bias, ps);
  gemm_wmma_kernel<<<MTOT / 64, 128, 0, stream>>>(ents, W, ps, p1ps, p2);
  bcast_kernel<<<MTOT, 256, 0, stream>>>(p1ps, p2, out);
}